// TrainableODE_73589969649865
// MI455X (gfx1250) — compile-verified
//
#include <hip/hip_runtime.h>
#include <stdint.h>
#include <math.h>

// ---------------------------------------------------------------------------
// Smith CVS model + Tsit5, batch of independent systems, one system per lane.
//
// MI455X reasoning (compile-only): workload is a 510-step *serial* RK chain
// per system, 2048 systems => 64 wave32 waves total. Latency-bound, not
// roofline-bound (0.4 GFLOP, 12.6 MB out). So: everything in registers,
// reciprocal resistances precomputed, driver exp()s hoisted off the
// y-dependent chain (ILP for the TRANS unit), one wave per workgroup to
// spread across as many SIMDs as possible. WMMA intentionally unused: the
// only contraction is depth<=6 over lane-private data; cross-lane WMMA
// marshalling would cost more than the FMAs it replaces, on the critical
// path. CDNA5-specific paths exercised: async global->LDS DMA (ASYNCcnt)
// to stage the shared time grid, then ds_load reads of it in the hot loop.
// ---------------------------------------------------------------------------

#define MAX_T_LDS 1024  // supports up to 1024 save points in LDS (4 KB)

struct Pars {
  float EesLv, EesRv, Eao, Evc, Epa, Epu;
  float rRmv, rRav, rRsys, rRtc, rRpv, rRpul;
};

__device__ __forceinline__ float cardiac_driver(float t) {
  // e = exp(-80 * (mod(t, 0.75) - 0.375)^2)
  float tm = t - 0.75f * floorf(t / 0.75f);
  float x  = tm - 0.375f;
  return __expf(-80.0f * (x * x));
}

__device__ __forceinline__ void cvs_rhs_e(float e, const float* __restrict__ V,
                                          const Pars& p, float* __restrict__ d) {
  const float Vlv = V[0], Vao = V[1], Vvc = V[2], Vrv = V[3], Vpa = V[4], Vpu = V[5];
  const float om  = 1.0f - e;
  const float Plv = e * p.EesLv * Vlv + om * 0.05f * (__expf(0.5f * Vlv) - 1.0f);
  const float Prv = e * p.EesRv * Vrv + om * 0.05f * (__expf(0.5f * Vrv) - 1.0f);
  const float Pao = p.Eao * Vao, Pvc = p.Evc * Vvc, Ppa = p.Epa * Vpa, Ppu = p.Epu * Vpu;
  const float Qmv  = fmaxf(Ppu - Plv, 0.0f) * p.rRmv;
  const float Qav  = fmaxf(Plv - Pao, 0.0f) * p.rRav;
  const float Qsys = (Pao - Pvc) * p.rRsys;
  const float Qtc  = fmaxf(Pvc - Prv, 0.0f) * p.rRtc;
  const float Qpv  = fmaxf(Prv - Ppa, 0.0f) * p.rRpv;
  const float Qpul = (Ppa - Ppu) * p.rRpul;
  d[0] = Qmv  - Qav;  d[1] = Qav  - Qsys; d[2] = Qsys - Qtc;
  d[3] = Qtc  - Qpv;  d[4] = Qpv  - Qpul; d[5] = Qpul - Qmv;
}

__device__ __forceinline__ void tsit5_step(float t, float dt, float* __restrict__ y,
                                           const Pars& p) {
  constexpr float A21 = 0.161f;
  constexpr float A31 = -0.008480655492356989f, A32 = 0.335480655492357f;
  constexpr float A41 = 2.8971530571054935f,  A42 = -6.359448489975075f, A43 = 4.3622954328695815f;
  constexpr float A51 = 5.325864828439257f,   A52 = -11.748883564062828f,
                  A53 = 7.4955393428898365f,  A54 = -0.09249506636175525f;
  constexpr float A61 = 5.86145544294642f,    A62 = -12.92096931784711f,
                  A63 = 8.159367898576159f,   A64 = -0.071584973281401f,
                  A65 = -0.028269050394068383f;
  constexpr float B1 = 0.09646076681806523f,  B2 = 0.01f, B3 = 0.4798896504144996f,
                  B4 = 1.379008574103742f,    B5 = -3.290069515436081f, B6 = 2.324710524099774f;

  // Six independent driver evaluations: y-independent, so these TRANS chains
  // overlap with the y-dependent VALU chain (CDNA5 VALU/TRANS co-execution).
  const float e1 = cardiac_driver(t);
  const float e2 = cardiac_driver(t + 0.161f * dt);
  const float e3 = cardiac_driver(t + 0.327f * dt);
  const float e4 = cardiac_driver(t + 0.9f * dt);
  const float e5 = cardiac_driver(t + 0.9800255409045097f * dt);
  const float e6 = cardiac_driver(t + dt);

  float k1[6], k2[6], k3[6], k4[6], k5[6], k6[6], yt[6];

  cvs_rhs_e(e1, y, p, k1);
#pragma unroll
  for (int i = 0; i < 6; ++i) yt[i] = fmaf(dt, A21 * k1[i], y[i]);
  cvs_rhs_e(e2, yt, p, k2);
#pragma unroll
  for (int i = 0; i < 6; ++i)
    yt[i] = fmaf(dt, fmaf(A32, k2[i], A31 * k1[i]), y[i]);
  cvs_rhs_e(e3, yt, p, k3);
#pragma unroll
  for (int i = 0; i < 6; ++i)
    yt[i] = fmaf(dt, fmaf(A43, k3[i], fmaf(A42, k2[i], A41 * k1[i])), y[i]);
  cvs_rhs_e(e4, yt, p, k4);
#pragma unroll
  for (int i = 0; i < 6; ++i)
    yt[i] = fmaf(dt, fmaf(A54, k4[i], fmaf(A53, k3[i], fmaf(A52, k2[i], A51 * k1[i]))), y[i]);
  cvs_rhs_e(e5, yt, p, k5);
#pragma unroll
  for (int i = 0; i < 6; ++i)
    yt[i] = fmaf(dt, fmaf(A65, k5[i], fmaf(A64, k4[i],
                    fmaf(A63, k3[i], fmaf(A62, k2[i], A61 * k1[i])))), y[i]);
  cvs_rhs_e(e6, yt, p, k6);
#pragma unroll
  for (int i = 0; i < 6; ++i) {
    float acc = fmaf(B6, k6[i], fmaf(B5, k5[i], fmaf(B4, k4[i],
                  fmaf(B3, k3[i], fmaf(B2, k2[i], B1 * k1[i])))));
    y[i] = fmaf(dt, acc, y[i]);
  }
}

// Integration loop. USE_LDS picks the time-grid source at compile time so the
// LDS variant references the __shared__ array directly (InferAddressSpaces ->
// ds_load_b64 broadcasts) and the fallback reads the uniform global pointer
// (scalarizable -> s_load). A runtime pointer select would force flat loads.
template <bool USE_LDS>
__device__ __forceinline__ void integrate(const float (&s_ts)[MAX_T_LDS],
                                          const float* __restrict__ ts,
                                          float* __restrict__ y, const Pars& p,
                                          float* __restrict__ out, int B, int T, int b) {
  for (int k = 0; k < T - 1; ++k) {
    float t0, t1;
    if (USE_LDS) { t0 = s_ts[k]; t1 = s_ts[k + 1]; }
    else         { t0 = ts[k];   t1 = ts[k + 1];   }
    const float dt = (t1 - t0) * 0.5f;  // N_SUB = 2 (exact)
    tsit5_step(t0,      dt, y, p);      // j = 0
    tsit5_step(t0 + dt, dt, y, p);      // j = 1
    float2* o2 = (float2*)(out + ((size_t)(k + 1) * B + b) * 6);
    o2[0] = make_float2(y[0], y[1]);
    o2[1] = make_float2(y[2], y[3]);
    o2[2] = make_float2(y[4], y[5]);
  }
}

__global__ __launch_bounds__(32)
void cvs_tsit5_kernel(const float* __restrict__ ts, const float* __restrict__ y0,
                      const float* __restrict__ params, float* __restrict__ out,
                      int B, int T) {
  __shared__ float s_ts[MAX_T_LDS];
  const int lane = threadIdx.x;  // blockDim.x == 32 (one wave32)

  // ---- Stage the wave-uniform time grid into LDS with CDNA5 async DMA ----
  const int bytes       = (T <= MAX_T_LDS ? T : 0) * 4;  // skip staging if too large
  const unsigned lds0   = (unsigned)(uintptr_t)&s_ts[0];
  const int b128_bytes  = bytes & ~15;
  for (int off = lane * 16; off < b128_bytes; off += 32 * 16) {
    unsigned    laddr = lds0 + (unsigned)off;
    const char* gaddr = (const char*)ts + off;
    asm volatile("global_load_async_to_lds_b128 %0, %1, off"
                 :: "v"(laddr), "v"(gaddr) : "memory");
  }
  for (int off = b128_bytes + lane * 4; off < bytes; off += 32 * 4) {
    unsigned    laddr = lds0 + (unsigned)off;
    const char* gaddr = (const char*)ts + off;
    asm volatile("global_load_async_to_lds_b32 %0, %1, off"
                 :: "v"(laddr), "v"(gaddr) : "memory");
  }
  asm volatile("s_wait_asynccnt 0x0" ::: "memory");
  __syncthreads();

  const int b = blockIdx.x * 32 + lane;
  if (b >= B) return;

  // ---- Per-system state fully register resident ----
  float y[6];
  const float* yp = y0 + (size_t)b * 6;
#pragma unroll
  for (int i = 0; i < 6; ++i) y[i] = yp[i];

  Pars p;
  const float* pp = params + (size_t)b * 12;
  p.EesLv = pp[0]; p.EesRv = pp[1]; p.Eao = pp[2];
  p.Evc   = pp[3]; p.Epa   = pp[4]; p.Epu = pp[5];
  p.rRmv  = 1.0f / pp[6];  p.rRav  = 1.0f / pp[7];  p.rRsys = 1.0f / pp[8];
  p.rRtc  = 1.0f / pp[9];  p.rRpv  = 1.0f / pp[10]; p.rRpul = 1.0f / pp[11];

  // Save point 0 = y0.  b*6*4 bytes is a multiple of 8 -> b64 stores are legal.
  {
    float2* o2 = (float2*)(out + (size_t)b * 6);
    o2[0] = make_float2(y[0], y[1]);
    o2[1] = make_float2(y[2], y[3]);
    o2[2] = make_float2(y[4], y[5]);
  }

  if (T <= MAX_T_LDS) integrate<true >(s_ts, ts, y, p, out, B, T, b);
  else                integrate<false>(s_ts, ts, y, p, out, B, T, b);
}

extern "C" void kernel_launch(void* const* d_in, const int* in_sizes, int n_in,
                              void* d_out, int out_size, void* d_ws, size_t ws_size,
                              hipStream_t stream) {
  const float* ts     = (const float*)d_in[0];
  const float* y0     = (const float*)d_in[1];
  const float* params = (const float*)d_in[2];
  float*       out    = (float*)d_out;

  const int T = in_sizes[0];       // 256 save points
  const int B = in_sizes[1] / 6;   // 2048 systems

  // One wave32 per workgroup: 64 independent workgroups spread across WGPs,
  // each wave alone on its SIMD with full issue bandwidth (latency-bound).
  dim3 block(32, 1, 1);
  dim3 grid((B + 31) / 32, 1, 1);
  cvs_tsit5_kernel<<<grid, block, 0, stream>>>(ts, y0, params, out, B, T);
}